// hc_kernel_loss_21749714387065
// MI455X (gfx1250) — compile-verified
//
#include <hip/hip_runtime.h>
#include <stdint.h>

#define B_  4
#define NO_ 8192
#define NS_ 30000
#define P_  2048
#define S_  10000
#define D_  64
#define TH_POS_    0.2f
#define TH_NEG_    10.0f
#define TH_KERNEL_ 0.15f

typedef float v2f __attribute__((ext_vector_type(2)));
typedef float v8f __attribute__((ext_vector_type(8)));

// ---------------- init per-row min arrays to +inf ----------------
__global__ void init_min_kernel(unsigned* rmA, unsigned* rmB, int n) {
  int i = blockIdx.x * blockDim.x + threadIdx.x;
  if (i < n) { rmA[i] = 0x7F800000u; rmB[i] = 0x7F800000u; }
}

// ---------------- gather features/coords + squared norms ----------------
// 16 threads per row (coalesced float4 per lane). idx==nullptr -> identity.
__global__ __launch_bounds__(256)
void gather_kernel(const float* __restrict__ srcF, const float* __restrict__ srcC,
                   const int* __restrict__ idx,
                   float* __restrict__ dstF, float* __restrict__ dstC,
                   float* __restrict__ dstN,
                   int rowsPerBatch, int srcPtsPerBatch, int totalRows) {
  int t = blockIdx.x * blockDim.x + threadIdx.x;
  int row  = t >> 4;
  int lane = t & 15;
  if (row >= totalRows) return;
  int b = row / rowsPerBatch;
  int i = row - b * rowsPerBatch;
  int s = idx ? idx[row] : i;
  size_t srcRow = (size_t)b * srcPtsPerBatch + (size_t)s;

  float4 v = *(const float4*)(srcF + srcRow * D_ + lane * 4);
  if (dstF) *(float4*)(dstF + (size_t)row * D_ + lane * 4) = v;
  float n = v.x * v.x + v.y * v.y + v.z * v.z + v.w * v.w;
#pragma unroll
  for (int m = 1; m < 16; m <<= 1) n += __shfl_xor(n, m, 32);
  if (lane == 0) dstN[row] = n;
  if (dstC && lane < 3) dstC[(size_t)row * 3 + lane] = srcC[srcRow * 3 + lane];
}

// ---------------- positive loss: per-row ||f0-f1||, block partials ----------------
__global__ __launch_bounds__(256)
void pos_kernel(const float* __restrict__ fObj, const float* __restrict__ fSc,
                float* __restrict__ part) {
  int gid = blockIdx.x * blockDim.x + threadIdx.x;   // 0..B*P-1
  const float4* a = (const float4*)(fObj + (size_t)gid * D_);
  const float4* b = (const float4*)(fSc  + (size_t)gid * D_);
  float d2 = 0.f;
#pragma unroll
  for (int i = 0; i < 16; ++i) {
    float4 x = a[i], y = b[i];
    float dx = x.x - y.x, dy = x.y - y.y, dz = x.z - y.z, dw = x.w - y.w;
    d2 += dx * dx + dy * dy + dz * dz + dw * dw;
  }
  float d = sqrtf(fmaxf(d2, 1e-12f));
  float r = fmaxf(d - TH_POS_, 0.0f);
  float loss = r * r;

  __shared__ float sl[256], sd[256];
  sl[threadIdx.x] = loss; sd[threadIdx.x] = d;
  __syncthreads();
  for (int s = 128; s > 0; s >>= 1) {
    if (threadIdx.x < s) {
      sl[threadIdx.x] += sl[threadIdx.x + s];
      sd[threadIdx.x] += sd[threadIdx.x + s];
    }
    __syncthreads();
  }
  if (threadIdx.x == 0) {
    part[2 * blockIdx.x]     = sl[0];
    part[2 * blockIdx.x + 1] = sd[0];
  }
}

// ---------------- hardest-negative: WMMA f32 16x16x4 distance tiles ----------------
// block = 128 threads (4 wave32). wave w owns rows [row0, row0+16).
// grid: (colChunks, P/64, B). Per-row running min merged via atomicMin(uint).
__global__ __launch_bounds__(128)
void neg_kernel(const float* __restrict__ subF, const float* __restrict__ subC,
                const float* __restrict__ subN,
                const float* __restrict__ candF, const float* __restrict__ candC,
                const float* __restrict__ candN,
                const float* __restrict__ diam,
                unsigned* __restrict__ rowmin,
                int M, int tilesTotal) {
  __shared__ float ldsB[16 * 68];   // candidate tile [16 rows][64 K], stride 68 (bank-conflict pad)
  __shared__ float ldsCd[48];       // candidate coords (16 x 3)
  __shared__ float ldsN[16];        // candidate squared norms

  int b = blockIdx.z;
  const float* subFb  = subF  + (size_t)b * P_ * D_;
  const float* subCb  = subC  + (size_t)b * P_ * 3;
  const float* subNb  = subN  + (size_t)b * P_;
  const float* candFb = candF + (size_t)b * M * D_;
  const float* candCb = candC + (size_t)b * M * 3;
  const float* candNb = candN + (size_t)b * M;
  unsigned* rowminB = rowmin + b * P_;
  float thresh = TH_KERNEL_ * diam[b];

  int tid  = threadIdx.x;
  int wave = tid >> 5;
  int lane = tid & 31;
  int nIdx = lane & 15;   // matrix row/col within tile for A/B frags and C column
  int hi   = lane >> 4;
  int row0 = (blockIdx.y * 4 + wave) * 16;

  // resident A fragments: 16 rows x 64 K fp32 (v2f per k-step; K = 4*kk + 2*hi .. +1)
  v2f af[16];
  {
    const float* arow = subFb + (size_t)(row0 + nIdx) * D_ + 2 * hi;
#pragma unroll
    for (int kk = 0; kk < 16; ++kk) af[kk] = *(const v2f*)(arow + kk * 4);
  }
  float rn[8], rcx[8], rcy[8], rcz[8], runmin[8];
#pragma unroll
  for (int v = 0; v < 8; ++v) {
    int rr = row0 + v + 8 * hi;        // C layout: VGPR v -> M = v (lanes 0-15), v+8 (lanes 16-31)
    rn[v]  = subNb[rr];
    rcx[v] = subCb[(size_t)rr * 3 + 0];
    rcy[v] = subCb[(size_t)rr * 3 + 1];
    rcz[v] = subCb[(size_t)rr * 3 + 2];
    runmin[v] = 3.4e38f;
  }

  int tStart = blockIdx.x * 64;
  int tEnd   = tStart + 64; if (tEnd > tilesTotal) tEnd = tilesTotal;

  for (int t = tStart; t < tEnd; ++t) {
    int col0 = t * 16;
    __syncthreads();
    {   // cooperative stage of candidate tile (16x64 floats) + coords + norms
      int r   = tid >> 3;
      int off = (tid & 7) * 8;
      const float* sp = candFb + (size_t)(col0 + r) * D_ + off;
      float4 p0 = *(const float4*)sp;
      float4 p1 = *(const float4*)(sp + 4);
      *(float4*)&ldsB[r * 68 + off]     = p0;
      *(float4*)&ldsB[r * 68 + off + 4] = p1;
      if (tid < 16) ldsN[tid]  = candNb[col0 + tid];
      if (tid < 48) ldsCd[tid] = candCb[(size_t)col0 * 3 + tid];
    }
    __syncthreads();

    float cn  = ldsN[nIdx];
    float ccx = ldsCd[nIdx * 3 + 0];
    float ccy = ldsCd[nIdx * 3 + 1];
    float ccz = ldsCd[nIdx * 3 + 2];

    v8f c = {};
#pragma unroll
    for (int kk = 0; kk < 16; ++kk) {
      v2f bf = *(const v2f*)&ldsB[nIdx * 68 + kk * 4 + 2 * hi];
      c = __builtin_amdgcn_wmma_f32_16x16x4_f32(false, af[kk], false, bf,
                                                (short)0, c, false, false);
    }
#pragma unroll
    for (int v = 0; v < 8; ++v) {
      float fd2  = rn[v] + cn - 2.0f * c[v];
      float feat = sqrtf(fmaxf(fd2, 1e-12f));
      float dx = rcx[v] - ccx, dy = rcy[v] - ccy, dz = rcz[v] - ccz;
      float geom = sqrtf(fmaxf(dx * dx + dy * dy + dz * dz, 1e-12f));
      float masked = feat + 1.0e6f * fmaxf(thresh - geom, 0.0f);
      runmin[v] = fminf(runmin[v], masked);
    }
  }

  // per-row min across the 16 lanes (columns), then global merge
#pragma unroll
  for (int v = 0; v < 8; ++v) {
    float m = runmin[v];
    m = fminf(m, __shfl_xor(m, 1, 32));
    m = fminf(m, __shfl_xor(m, 2, 32));
    m = fminf(m, __shfl_xor(m, 4, 32));
    m = fminf(m, __shfl_xor(m, 8, 32));
    if (nIdx == 0)
      atomicMin(&rowminB[row0 + v + 8 * hi], __float_as_uint(m));
  }
}

// ---------------- final reduction to the 6 outputs ----------------
__global__ __launch_bounds__(256)
void final_kernel(const unsigned* __restrict__ rmSc, const unsigned* __restrict__ rmOb,
                  const float* __restrict__ posPart, float* __restrict__ out) {
  float acc[6] = {0, 0, 0, 0, 0, 0};
  for (int i = threadIdx.x; i < B_ * P_; i += 256) {
    float d = __uint_as_float(rmOb[i]);
    float r = fmaxf(TH_NEG_ - d, 0.0f);
    acc[1] += r * r; acc[4] += d;
    d = __uint_as_float(rmSc[i]);
    r = fmaxf(TH_NEG_ - d, 0.0f);
    acc[2] += r * r; acc[5] += d;
  }
  if (threadIdx.x < 32) {               // 32 pos blocks
    acc[0] = posPart[2 * threadIdx.x];
    acc[3] = posPart[2 * threadIdx.x + 1];
  }
  __shared__ float red[6][256];
  for (int j = 0; j < 6; ++j) red[j][threadIdx.x] = acc[j];
  __syncthreads();
  for (int s = 128; s > 0; s >>= 1) {
    if (threadIdx.x < s)
      for (int j = 0; j < 6; ++j) red[j][threadIdx.x] += red[j][threadIdx.x + s];
    __syncthreads();
  }
  if (threadIdx.x == 0) {
    const float inv = 1.0f / (float)(B_ * P_);
    for (int j = 0; j < 6; ++j) out[j] = red[j][0] * inv;
  }
}

// ---------------- launch ----------------
extern "C" void kernel_launch(void* const* d_in, const int* in_sizes, int n_in,
                              void* d_out, int out_size, void* d_ws, size_t ws_size,
                              hipStream_t stream) {
  (void)in_sizes; (void)n_in; (void)out_size; (void)ws_size;
  const float* obj_coords   = (const float*)d_in[0];
  const float* obj_feats    = (const float*)d_in[1];
  const float* scene_coords = (const float*)d_in[2];
  const float* scene_feats  = (const float*)d_in[3];
  const float* obj_diam     = (const float*)d_in[4];
  const int*   pio          = (const int*)d_in[5];
  const int*   pis          = (const int*)d_in[6];
  const int*   si           = (const int*)d_in[7];
  float* out = (float*)d_out;

  char* w = (char*)d_ws;
  auto alloc = [&](size_t elems) {
    float* p = (float*)w;
    w += ((elems * 4 + 255) / 256) * 256;
    return p;
  };
  float*    subF_sc  = alloc((size_t)B_ * P_ * D_);   // sf[pis]
  float*    subF_ob  = alloc((size_t)B_ * P_ * D_);   // of[pio]
  float*    candF_sc = alloc((size_t)B_ * S_ * D_);   // sf[si]
  float*    subC_sc  = alloc((size_t)B_ * P_ * 3);
  float*    subC_ob  = alloc((size_t)B_ * P_ * 3);
  float*    candC_sc = alloc((size_t)B_ * S_ * 3);
  float*    subN_sc  = alloc((size_t)B_ * P_);
  float*    subN_ob  = alloc((size_t)B_ * P_);
  float*    candN_sc = alloc((size_t)B_ * S_);
  float*    objN     = alloc((size_t)B_ * NO_);       // norms of all obj pts
  unsigned* rm_sc    = (unsigned*)alloc((size_t)B_ * P_);
  unsigned* rm_ob    = (unsigned*)alloc((size_t)B_ * P_);
  float*    posPart  = alloc(64);

  init_min_kernel<<<(B_ * P_ + 255) / 256, 256, 0, stream>>>(rm_sc, rm_ob, B_ * P_);

  gather_kernel<<<(B_ * P_ * 16 + 255) / 256, 256, 0, stream>>>(
      scene_feats, scene_coords, pis, subF_sc, subC_sc, subN_sc, P_, NS_, B_ * P_);
  gather_kernel<<<(B_ * P_ * 16 + 255) / 256, 256, 0, stream>>>(
      obj_feats, obj_coords, pio, subF_ob, subC_ob, subN_ob, P_, NO_, B_ * P_);
  gather_kernel<<<(B_ * S_ * 16 + 255) / 256, 256, 0, stream>>>(
      scene_feats, scene_coords, si, candF_sc, candC_sc, candN_sc, S_, NS_, B_ * S_);
  gather_kernel<<<(B_ * NO_ * 16 + 255) / 256, 256, 0, stream>>>(
      obj_feats, nullptr, nullptr, nullptr, nullptr, objN, NO_, NO_, B_ * NO_);

  pos_kernel<<<(B_ * P_) / 256, 256, 0, stream>>>(subF_ob, subF_sc, posPart);

  // scene negatives: 625 column tiles -> 10 chunks of 64
  neg_kernel<<<dim3((S_ / 16 + 63) / 64, P_ / 64, B_), 128, 0, stream>>>(
      subF_sc, subC_sc, subN_sc, candF_sc, candC_sc, candN_sc, obj_diam,
      rm_sc, S_, S_ / 16);
  // object negatives: 512 column tiles -> 8 chunks of 64
  neg_kernel<<<dim3((NO_ / 16 + 63) / 64, P_ / 64, B_), 128, 0, stream>>>(
      subF_ob, subC_ob, subN_ob, obj_feats, obj_coords, objN, obj_diam,
      rm_ob, NO_, NO_ / 16);

  final_kernel<<<1, 256, 0, stream>>>(rm_sc, rm_ob, posPart, out);
}